// GatedDeltaRuleWithProjectionModel_65017214926852
// MI455X (gfx1250) — compile-verified
//
#include <hip/hip_runtime.h>
#include <hip/hip_bf16.h>
#include <math.h>

// Problem constants (from reference): B=2, T=2048, NH=16, HD=128, VD=256
// KD = 2048, VDIM = 4096, proj width O = 2*KD + VDIM + 2*NH = 8224
#define BSZ   2
#define TT    2048
#define NH    16
#define HD    128
#define VD    256
#define KD    2048
#define OW    8224            // proj row width
#define MROWS (BSZ * TT)      // 4096 GEMM rows
#define NPAN  (KD / 16)       // 128 K panels of 16

typedef __attribute__((ext_vector_type(2))) float v2f;
typedef __attribute__((ext_vector_type(8))) float v8f;

// ---------------------------------------------------------------------------
// Kernel 1: proj = x @ W^T using V_WMMA_F32_16X16X4_F32.
// Block = 256 threads (8 wave32) computes 128(M) x 128(N).
// Wave tile = 32(M) x 64(N)  -> 2x4 v8f accumulators (8 WMMA per k-step,
// 6 LDS b64 fragment loads per k-step).
// K staged through double-buffered LDS panels of 16 (1 barrier per panel).
// LDS row stride = 18 floats: kills the stride-16 bank conflict while
// keeping every b64 fragment read 8-byte aligned.
// ---------------------------------------------------------------------------
#define LDSX 18

__global__ __launch_bounds__(256) void proj_gemm_wmma(
    const float* __restrict__ x,   // [4096, 2048]
    const float* __restrict__ W,   // [8224, 2048]
    float* __restrict__ proj)      // [4096, 8224]
{
    __shared__ float xs[2][128 * LDSX];   // x tile  [128][16] padded, x2 buf
    __shared__ float wl[2][128 * LDSX];   // W tile  [128][16] padded, x2 buf

    const int tid  = threadIdx.x;
    const int lane = tid & 31;
    const int wave = tid >> 5;
    const int wm   = wave & 3;     // 0..3 -> M offset wm*32
    const int wn   = wave >> 2;    // 0..1 -> N offset wn*64
    const int lr   = lane & 15;    // row-in-fragment
    const int lh   = lane >> 4;    // half select (k pair)

    const int m0 = blockIdx.y * 128;
    const int n0 = blockIdx.x * 128;

    // staging coordinates (128 rows x 16 cols = 512 float4 / 256 thr = 2 each)
    const int srow0 = tid >> 2;            // rows 0..63
    const int srow1 = (tid + 256) >> 2;    // rows 64..127
    const int sc4   = (tid & 3) * 4;       // col group
    int wrow0 = n0 + srow0; if (wrow0 > OW - 1) wrow0 = OW - 1;
    int wrow1 = n0 + srow1; if (wrow1 > OW - 1) wrow1 = OW - 1;

    v8f acc[2][4];
#pragma unroll
    for (int i = 0; i < 2; ++i)
#pragma unroll
        for (int j = 0; j < 4; ++j)
            acc[i][j] = (v8f)(0.0f);

    float4 xr0, xr1, wr0, wr1;

    // prologue: fetch + stage panel 0
    xr0 = *(const float4*)(x + (size_t)(m0 + srow0) * KD + sc4);
    xr1 = *(const float4*)(x + (size_t)(m0 + srow1) * KD + sc4);
    wr0 = *(const float4*)(W + (size_t)wrow0 * KD + sc4);
    wr1 = *(const float4*)(W + (size_t)wrow1 * KD + sc4);
    {
        float* d0 = &xs[0][srow0 * LDSX + sc4];
        d0[0] = xr0.x; d0[1] = xr0.y; d0[2] = xr0.z; d0[3] = xr0.w;
        float* d1 = &xs[0][srow1 * LDSX + sc4];
        d1[0] = xr1.x; d1[1] = xr1.y; d1[2] = xr1.z; d1[3] = xr1.w;
        float* e0 = &wl[0][srow0 * LDSX + sc4];
        e0[0] = wr0.x; e0[1] = wr0.y; e0[2] = wr0.z; e0[3] = wr0.w;
        float* e1 = &wl[0][srow1 * LDSX + sc4];
        e1[0] = wr1.x; e1[1] = wr1.y; e1[2] = wr1.z; e1[3] = wr1.w;
    }
    __syncthreads();

    for (int p = 0; p < NPAN; ++p) {
        const int buf = p & 1;
        const int k1  = (p + 1) * 16;

        // fetch next panel (overlaps with WMMA below; tracked by LOADcnt)
        if (p + 1 < NPAN) {
            xr0 = *(const float4*)(x + (size_t)(m0 + srow0) * KD + k1 + sc4);
            xr1 = *(const float4*)(x + (size_t)(m0 + srow1) * KD + k1 + sc4);
            wr0 = *(const float4*)(W + (size_t)wrow0 * KD + k1 + sc4);
            wr1 = *(const float4*)(W + (size_t)wrow1 * KD + k1 + sc4);
        }

        // compute on current buffer: 4 wmma k-steps, 8 WMMA each
        const float* xb = xs[buf];
        const float* wb = wl[buf];
#pragma unroll
        for (int kk = 0; kk < 16; kk += 4) {
            const v2f a0 = *(const v2f*)&xb[(wm * 32 +  0 + lr) * LDSX + kk + 2 * lh];
            const v2f a1 = *(const v2f*)&xb[(wm * 32 + 16 + lr) * LDSX + kk + 2 * lh];
            const v2f b0 = *(const v2f*)&wb[(wn * 64 +  0 + lr) * LDSX + kk + 2 * lh];
            const v2f b1 = *(const v2f*)&wb[(wn * 64 + 16 + lr) * LDSX + kk + 2 * lh];
            const v2f b2 = *(const v2f*)&wb[(wn * 64 + 32 + lr) * LDSX + kk + 2 * lh];
            const v2f b3 = *(const v2f*)&wb[(wn * 64 + 48 + lr) * LDSX + kk + 2 * lh];

            acc[0][0] = __builtin_amdgcn_wmma_f32_16x16x4_f32(
                false, a0, false, b0, (short)0, acc[0][0], false, false);
            acc[0][1] = __builtin_amdgcn_wmma_f32_16x16x4_f32(
                false, a0, false, b1, (short)0, acc[0][1], false, false);
            acc[0][2] = __builtin_amdgcn_wmma_f32_16x16x4_f32(
                false, a0, false, b2, (short)0, acc[0][2], false, false);
            acc[0][3] = __builtin_amdgcn_wmma_f32_16x16x4_f32(
                false, a0, false, b3, (short)0, acc[0][3], false, false);
            acc[1][0] = __builtin_amdgcn_wmma_f32_16x16x4_f32(
                false, a1, false, b0, (short)0, acc[1][0], false, false);
            acc[1][1] = __builtin_amdgcn_wmma_f32_16x16x4_f32(
                false, a1, false, b1, (short)0, acc[1][1], false, false);
            acc[1][2] = __builtin_amdgcn_wmma_f32_16x16x4_f32(
                false, a1, false, b2, (short)0, acc[1][2], false, false);
            acc[1][3] = __builtin_amdgcn_wmma_f32_16x16x4_f32(
                false, a1, false, b3, (short)0, acc[1][3], false, false);
        }

        // stage next panel into the other buffer (its previous readers
        // finished before the barrier that ended iteration p-1)
        if (p + 1 < NPAN) {
            float* d0 = &xs[buf ^ 1][srow0 * LDSX + sc4];
            d0[0] = xr0.x; d0[1] = xr0.y; d0[2] = xr0.z; d0[3] = xr0.w;
            float* d1 = &xs[buf ^ 1][srow1 * LDSX + sc4];
            d1[0] = xr1.x; d1[1] = xr1.y; d1[2] = xr1.z; d1[3] = xr1.w;
            float* e0 = &wl[buf ^ 1][srow0 * LDSX + sc4];
            e0[0] = wr0.x; e0[1] = wr0.y; e0[2] = wr0.z; e0[3] = wr0.w;
            float* e1 = &wl[buf ^ 1][srow1 * LDSX + sc4];
            e1[0] = wr1.x; e1[1] = wr1.y; e1[2] = wr1.z; e1[3] = wr1.w;
        }
        __syncthreads();   // single barrier per panel
    }

    // ---- store: C/D layout lane l, vgpr r -> M = r + 8*(l>>4), N = l&15
#pragma unroll
    for (int i = 0; i < 2; ++i) {
#pragma unroll
        for (int j = 0; j < 4; ++j) {
            const int col = n0 + wn * 64 + j * 16 + lr;
            if (col < OW) {
                const int rbase = m0 + wm * 32 + i * 16 + 8 * lh;
#pragma unroll
                for (int r = 0; r < 8; ++r)
                    proj[(size_t)(rbase + r) * OW + col] = acc[i][j][r];
            }
        }
    }
}

// ---------------------------------------------------------------------------
// Kernel 2: in-place L2 normalization of q and k head vectors (len 128).
// grid = B*T*2*NH blocks of 128 threads; wave shuffle + LDS reduction.
// ---------------------------------------------------------------------------
__global__ __launch_bounds__(128) void l2norm_qk(float* __restrict__ proj)
{
    const int vec   = blockIdx.x;       // 0 .. B*T*2*NH-1
    const int row   = vec >> 5;         // / (2*NH)
    const int rem   = vec & 31;
    const int which = rem >> 4;         // 0 = q block, 1 = k block
    const int head  = rem & 15;
    const size_t base = (size_t)row * OW + which * KD + head * HD;

    const float val = proj[base + threadIdx.x];
    float p = val * val;
#pragma unroll
    for (int off = 16; off > 0; off >>= 1)
        p += __shfl_xor(p, off, 32);

    __shared__ float sred[4];
    const int lane = threadIdx.x & 31;
    const int wv   = threadIdx.x >> 5;
    if (lane == 0) sred[wv] = p;
    __syncthreads();
    const float tot = sred[0] + sred[1] + sred[2] + sred[3];
    const float inv = 1.0f / fmaxf(sqrtf(tot), 1e-12f);
    proj[base + threadIdx.x] = val * inv;
}

// ---------------------------------------------------------------------------
// Kernel 3: sequential gated delta-rule scan.
// One workgroup per (b, head, v-slice of 64): 128 WGs, 256 threads each.
// Threads = (hg 0..3) x (v 0..63); state slice S[h=128][v=64] in LDS (32 KB).
// Per step: 32-iteration inner loops + small LDS tree reductions.
// ---------------------------------------------------------------------------
__global__ __launch_bounds__(256) void delta_scan(
    const float* __restrict__ proj,    // [4096, 8224] (q,k normalized)
    const float* __restrict__ state0,  // [B, NH, VD, HD]
    float* __restrict__ out)           // [B, T, NH, VD]
{
    __shared__ float S[HD * 64];       // S[h][v_local], stride 64
    __shared__ float kbuf[HD];
    __shared__ float qbuf[HD];
    __shared__ float vbuf[64];
    __shared__ float red[2 * 4 * 64];  // [which][hg][v]

    const int tid = threadIdx.x;
    const int hg  = tid >> 6;          // 0..3 -> h range hg*32..+32
    const int v   = tid & 63;          // local v column
    const int wg  = blockIdx.x;        // 0..127
    const int vs  = wg & 3;
    const int n   = (wg >> 2) & 15;
    const int b   = wg >> 6;
    const int vbase = vs * 64;
    const int hb = hg * 32;

    // init state slice from input state: state0[((b*NH+n)*VD + vg)*HD + h]
    const size_t sb = ((size_t)(b * NH + n) * VD + vbase + v) * HD;
#pragma unroll
    for (int i = 0; i < 32; ++i) {
        const int h = hb + i;
        S[h * 64 + v] = state0[sb + h];
    }
    __syncthreads();

    for (int t = 0; t < TT; ++t) {
        const float* pr = proj + (size_t)(b * TT + t) * OW;
        if (tid < HD)       kbuf[tid]       = pr[KD + n * HD + tid];
        else                qbuf[tid - HD]  = pr[n * HD + (tid - HD)];
        if (tid < 64)       vbuf[tid]       = pr[2 * KD + n * VD + vbase + tid];
        const float g   = 1.0f / (1.0f + __expf(-pr[2 * KD + NH * VD + n]));
        const float bet = 1.0f / (1.0f + __expf(-pr[2 * KD + NH * VD + NH + n]));
        __syncthreads();

        // partial Sk over this thread's h range
        float psk = 0.0f;
#pragma unroll 8
        for (int i = 0; i < 32; ++i)
            psk += S[(hb + i) * 64 + v] * kbuf[hb + i];
        red[hg * 64 + v] = psk;
        __syncthreads();

        const float Sk = red[v] + red[64 + v] + red[128 + v] + red[192 + v];
        const float delta = bet * (vbuf[v] - g * Sk);

        // state update + partial output
        float po = 0.0f;
#pragma unroll 8
        for (int i = 0; i < 32; ++i) {
            const int h = hb + i;
            const float s = g * S[h * 64 + v] + delta * kbuf[h];
            S[h * 64 + v] = s;
            po += s * qbuf[h];
        }
        red[256 + hg * 64 + v] = po;
        __syncthreads();

        if (hg == 0) {
            const float o = red[256 + v] + red[320 + v] +
                            red[384 + v] + red[448 + v];
            out[((size_t)(b * TT + t) * NH + n) * VD + vbase + v] = o;
        }
        __syncthreads();   // protect kbuf/qbuf/vbuf/red for next step
    }
}

// ---------------------------------------------------------------------------
extern "C" void kernel_launch(void* const* d_in, const int* in_sizes, int n_in,
                              void* d_out, int out_size, void* d_ws, size_t ws_size,
                              hipStream_t stream)
{
    const float* x      = (const float*)d_in[0];   // [B,T,KD]
    const float* W      = (const float*)d_in[1];   // [8224, 2048]
    const float* state0 = (const float*)d_in[2];   // [B,NH,VD,HD]
    float*       out    = (float*)d_out;           // [B,T,NH,VD]
    float*       proj   = (float*)d_ws;            // [4096, 8224] f32 (~128.5 MiB)

    (void)in_sizes; (void)n_in; (void)out_size; (void)ws_size;

    // 1) projection GEMM: grid covers N=8224 (65 blocks of 128, edge masked)
    dim3 ggrid((OW + 127) / 128, MROWS / 128);
    proj_gemm_wmma<<<ggrid, 256, 0, stream>>>(x, W, proj);

    // 2) l2-normalize q and k vectors in place
    l2norm_qk<<<BSZ * TT * 2 * NH, 128, 0, stream>>>(proj);

    // 3) sequential gated delta scan
    delta_scan<<<BSZ * NH * 4, 256, 0, stream>>>(proj, state0, out);
}